// LearnedSolverNet_29480655519971
// MI455X (gfx1250) — compile-verified
//
#include <hip/hip_runtime.h>
#include <hip/hip_bf16.h>
#include <math.h>

typedef __attribute__((ext_vector_type(2))) float v2f;
typedef __attribute__((ext_vector_type(8))) float v8f;

#define BATCH   64
#define H_IN    240
#define W_IN    320
#define HW      (H_IN*W_IN)          // 76800
#define P1H     120
#define P1W     160
#define A1H     60
#define A1W     80
#define A2H     30
#define A2W     40
#define C1      8
#define C2      16
#define KFC     19200
#define KSPLIT  30
#define KCHUNK  640                   // 19200/30
#define KT      64
#define NCHUNK  64
#define CHW     (HW/NCHUNK)           // 1200 floats = 300 float4
#define CHW4    (CHW/4)               // 300
#define HW4     (HW/4)                // 19200

// ---------------- JtR partials (B128 vectorized): jtrp[b][chunk][i] ----------------
__global__ void k_jtr_partial(const float4* __restrict__ Jt4,
                              const float4* __restrict__ wts4,
                              const float4* __restrict__ Rres4,
                              float* __restrict__ jtrp) {
  int b  = blockIdx.x;
  int ch = blockIdx.y;
  int tid = threadIdx.x;
  const float4* wb = wts4  + (size_t)b*HW4 + (size_t)ch*CHW4;
  const float4* rb = Rres4 + (size_t)b*HW4 + (size_t)ch*CHW4;
  const float4* jb = Jt4   + (size_t)b*6*HW4 + (size_t)ch*CHW4;
  float acc[6] = {0.f,0.f,0.f,0.f,0.f,0.f};
  for (int t = tid; t < CHW4; t += 256) {
    float4 w4 = wb[t];
    float4 r4 = rb[t];
    float4 wr; wr.x=w4.x*r4.x; wr.y=w4.y*r4.y; wr.z=w4.z*r4.z; wr.w=w4.w*r4.w;
#pragma unroll
    for (int i = 0; i < 6; ++i) {
      float4 j4 = jb[(size_t)i*HW4 + t];
      acc[i] += j4.x*wr.x + j4.y*wr.y + j4.z*wr.z + j4.w*wr.w;
    }
  }
  __shared__ float red[6][256];
#pragma unroll
  for (int i=0;i<6;++i) red[i][tid]=acc[i];
  __syncthreads();
  for (int s=128;s>0;s>>=1){
    if (tid<s) {
#pragma unroll
      for(int i=0;i<6;++i) red[i][tid]+=red[i][tid+s];
    }
    __syncthreads();
  }
  if (tid<6) jtrp[((size_t)b*NCHUNK + ch)*6 + tid] = red[tid][0];
}

// ---------------- maxpool 240x320 -> 120x160 ----------------
__global__ void k_pool0(const float* __restrict__ Rres, float* __restrict__ p1) {
  int idx = blockIdx.x*256 + threadIdx.x;
  if (idx >= BATCH*P1H*P1W) return;
  int x = idx % P1W; int y = (idx / P1W) % P1H; int b = idx / (P1W*P1H);
  const float* src = Rres + (size_t)b*HW + (size_t)(2*y)*W_IN + 2*x;
  p1[idx] = fmaxf(fmaxf(src[0], src[1]), fmaxf(src[W_IN], src[W_IN+1]));
}

// ---------------- conv1(1->8,3x3,pad1)+relu+pool -> a1[b][c][60][80] ----------------
__global__ void k_conv1pool(const float* __restrict__ p1,
                            const float* __restrict__ w1,
                            const float* __restrict__ b1,
                            float* __restrict__ a1) {
  int idx = blockIdx.x*256 + threadIdx.x;
  if (idx >= BATCH*C1*A1H*A1W) return;
  int x = idx % A1W; int y=(idx/A1W)%A1H; int c=(idx/(A1W*A1H))%C1; int b=idx/(A1W*A1H*C1);
  float wk[9];
#pragma unroll
  for (int i=0;i<9;++i) wk[i]=w1[c*9+i];
  float bias = b1[c];
  const float* src = p1 + (size_t)b*P1H*P1W;
  float m = -INFINITY;
#pragma unroll
  for (int dy=0; dy<2; ++dy)
#pragma unroll
  for (int dx=0; dx<2; ++dx) {
    int Y = 2*y+dy, X = 2*x+dx;
    float s = bias;
#pragma unroll
    for (int ky=0; ky<3; ++ky) {
      int yy = Y-1+ky;
      if (yy<0||yy>=P1H) continue;
#pragma unroll
      for (int kx=0; kx<3; ++kx) {
        int xx = X-1+kx;
        if (xx<0||xx>=P1W) continue;
        s += wk[ky*3+kx]*src[yy*P1W+xx];
      }
    }
    m = fmaxf(m, s);
  }
  a1[idx] = fmaxf(m, 0.f);   // relu(max) == max(relu) (monotone)
}

// ---------------- conv2(8->16,3x3,pad1)+relu+pool -> h[b][co*1200+y*40+x] ----------------
__global__ void k_conv2pool(const float* __restrict__ a1,
                            const float* __restrict__ w2,
                            const float* __restrict__ b2,
                            float* __restrict__ h) {
  int idx = blockIdx.x*256 + threadIdx.x;
  if (idx >= BATCH*C2*A2H*A2W) return;
  int x = idx % A2W; int y=(idx/A2W)%A2H; int co=(idx/(A2W*A2H))%C2; int b=idx/(A2W*A2H*C2);
  float bias=b2[co];
  float m=-INFINITY;
  const float* wco = w2 + co*C1*9;
  for (int dy=0;dy<2;++dy) for(int dx=0;dx<2;++dx){
    int Y=2*y+dy, X=2*x+dx;
    float s=bias;
    for (int ci=0; ci<C1; ++ci){
      const float* src = a1 + ((size_t)b*C1+ci)*(A1H*A1W);
      const float* wk = wco + ci*9;
#pragma unroll
      for(int ky=0;ky<3;++ky){
        int yy=Y-1+ky; if(yy<0||yy>=A1H) continue;
#pragma unroll
        for(int kx=0;kx<3;++kx){
          int xx=X-1+kx; if(xx<0||xx>=A1W) continue;
          s += wk[ky*3+kx]*src[yy*A1W+xx];
        }
      }
    }
    m=fmaxf(m,s);
  }
  h[(size_t)b*KFC + co*(A2H*A2W) + y*A2W + x] = fmaxf(m,0.f);
}

// ---------------- fc1 GEMM with FP32 WMMA 16x16x4, K split over blocks ----------------
// C[m][n] = sum_k h[m][k] * fc1_w[n][k];  partial per block in ws.
__global__ void __launch_bounds__(512)
k_fc1_wmma(const float* __restrict__ h,
           const float* __restrict__ fw,
           float* __restrict__ partial) {
  __shared__ float Ah[64][KT+4];   // row stride 68 floats (272B, 16B-aligned, conflict-free)
  __shared__ float Bw[64][KT+4];
  int tid  = threadIdx.x;
  int wave = tid >> 5;
  int lane = tid & 31;
  int half = lane >> 4;
  int l    = lane & 15;
  int m0 = (wave >> 2) * 16;
  int n0 = (wave & 3) * 16;
  int kbase0 = blockIdx.x * KCHUNK;
  v8f acc = {};
  for (int ko = 0; ko < KCHUNK; ko += KT) {
    int kb = kbase0 + ko;
    // cooperative B128 tile loads: 1024 float4 each / 512 thr = 2 iters, no divergence
    for (int idx = tid; idx < 64*(KT/4); idx += 512) {
      int row = idx >> 4;          // 16 float4 per row
      int col = (idx & 15) << 2;
      *(float4*)&Ah[row][col] = *(const float4*)&h [(size_t)row*KFC + kb + col];
      *(float4*)&Bw[row][col] = *(const float4*)&fw[(size_t)row*KFC + kb + col];
    }
    __syncthreads();
#pragma unroll
    for (int kk = 0; kk < KT; kk += 4) {
      // A-frag 16x4: lanes 0-15 rows, K={0,1} low half-wave, K={2,3} high half-wave
      v2f a, bb;
      a.x  = Ah[m0+l][kk + 2*half];
      a.y  = Ah[m0+l][kk + 2*half + 1];
      // B-frag 4x16 (B[k][n] = fc1_w[n][k]), mirrored layout
      bb.x = Bw[n0+l][kk + 2*half];
      bb.y = Bw[n0+l][kk + 2*half + 1];
      acc = __builtin_amdgcn_wmma_f32_16x16x4_f32(
          /*neg_a=*/false, a, /*neg_b=*/false, bb,
          /*c_mod=*/(short)0, acc, /*reuse_a=*/false, /*reuse_b=*/false);
    }
    __syncthreads();
  }
  float* outp = partial + (size_t)blockIdx.x*64*64;
#pragma unroll
  for (int r = 0; r < 8; ++r) {
    int m = m0 + half*8 + r;     // C/D layout: vgpr r -> M = r (+8 for high half-wave)
    int n = n0 + l;
    outp[m*64 + n] = acc[r];
  }
}

__global__ void k_fc1_reduce(const float* __restrict__ partial,
                             const float* __restrict__ fb,
                             float* __restrict__ Hfc) {
  int idx = blockIdx.x*256 + threadIdx.x;
  if (idx >= 64*64) return;
  float s = fb[idx & 63];
  for (int p = 0; p < KSPLIT; ++p) s += partial[(size_t)p*4096 + idx];
  Hfc[idx] = fmaxf(s, 0.f);
}

// ---------------- fc2 + softmax + @consts + sigmoid + 10^x -> lam[b][6] ----------------
__global__ void k_damp(const float* __restrict__ Hfc, const float* __restrict__ f2w,
                       const float* __restrict__ f2b, const float* __restrict__ consts,
                       float* __restrict__ lamb) {
  int b = threadIdx.x;
  if (b >= BATCH) return;
  float t[6];
  for (int j=0;j<6;++j){
    float s = f2b[j];
    for (int n=0;n<64;++n) s += Hfc[b*64+n]*f2w[j*64+n];
    t[j]=s;
  }
  float mx = t[0];
  for (int j=1;j<6;++j) mx=fmaxf(mx,t[j]);
  float se=0.f, e[6];
  for(int j=0;j<6;++j){ e[j]=expf(t[j]-mx); se+=e[j]; }
  for(int j=0;j<6;++j) e[j]/=se;
  for(int j=0;j<6;++j){
    float wc=0.f;
    for(int i=0;i<6;++i) wc += e[i]*consts[i*6+j];
    float sg = 1.f/(1.f+expf(-wc));
    lamb[b*6+j] = powf(10.f, -6.f + sg);   // min=-6, max=-5
  }
}

// ---------------- per-b 6x6 solve + Rodrigues + pose compose ----------------
__global__ void k_solve(const float* __restrict__ JtJ, const float* __restrict__ jtrp,
                        const float* __restrict__ lamb,
                        const float* __restrict__ poseR, const float* __restrict__ poset,
                        float* __restrict__ out) {
  int b = threadIdx.x;
  if (b >= BATCH) return;
  // finish JtR reduction
  float rhs[6];
  for(int i=0;i<6;++i){
    float s=0.f;
    for(int c=0;c<NCHUNK;++c) s += jtrp[((size_t)b*NCHUNK+c)*6 + i];
    rhs[i]=s;
  }
  const float* Jb = JtJ + b*36;
  float tr=0.f;
  for(int i=0;i<6;++i) tr += Jb[i*6+i];
  float Mm[6][7];
  for(int i=0;i<6;++i){
    for(int j=0;j<6;++j) Mm[i][j]=Jb[i*6+j];
    Mm[i][i] += lamb[b*6+i] + 1e-6f*tr + 1e-6f;   // damping + eps*trace + REG
    Mm[i][6] = rhs[i];
  }
  // Gauss-Jordan with partial pivoting (== inv(H) @ JtR numerically)
  for(int col=0; col<6; ++col){
    int piv=col; float pv=fabsf(Mm[col][col]);
    for(int r=col+1;r<6;++r){ float v=fabsf(Mm[r][col]); if(v>pv){pv=v;piv=r;} }
    if(piv!=col){ for(int j=0;j<7;++j){ float tmp=Mm[col][j]; Mm[col][j]=Mm[piv][j]; Mm[piv][j]=tmp; } }
    float inv = 1.f/Mm[col][col];
    for(int j=0;j<7;++j) Mm[col][j]*=inv;
    for(int r=0;r<6;++r){
      if(r==col) continue;
      float f=Mm[r][col];
      for(int j=0;j<7;++j) Mm[r][j]-=f*Mm[col][j];
    }
  }
  float xi[6]; for(int i=0;i<6;++i) xi[i]=Mm[i][6];
  // Rodrigues on w = -xi[0:3]
  float w0=-xi[0], w1=-xi[1], w2=-xi[2];
  float th = fmaxf(sqrtf(w0*w0+w1*w1+w2*w2), 1e-12f);
  float k0=w0/th, k1=w1/th, k2=w2/th;
  float s=sinf(th), c=1.f-cosf(th);
  float A[3][3] = {{0.f,-k2,k1},{k2,0.f,-k0},{-k1,k0,0.f}};
  float A2[3][3];
  for(int i=0;i<3;++i)for(int j=0;j<3;++j){
    float v=0.f; for(int m=0;m<3;++m) v+=A[i][m]*A[m][j];
    A2[i][j]=v;
  }
  float dR[3][3];
  for(int i=0;i<3;++i)for(int j=0;j<3;++j)
    dR[i][j] = (i==j?1.f:0.f) + s*A[i][j] + c*A2[i][j];
  float u0=xi[3],u1=xi[4],u2=xi[5];
  float dt[3];
  for(int i=0;i<3;++i) dt[i] = -(dR[i][0]*u0 + dR[i][1]*u1 + dR[i][2]*u2);
  const float* PR = poseR + b*9;
  const float* PT = poset + b*3;
  float* ob = out + b*12;
  for(int i=0;i<3;++i){
    float r0=0.f,r1=0.f,r2=0.f,tv=0.f;
    for(int m=0;m<3;++m){
      r0 += PR[i*3+m]*dR[m][0];
      r1 += PR[i*3+m]*dR[m][1];
      r2 += PR[i*3+m]*dR[m][2];
      tv += PR[i*3+m]*dt[m];
    }
    ob[i*4+0]=r0; ob[i*4+1]=r1; ob[i*4+2]=r2; ob[i*4+3]=tv+PT[i];
  }
}

extern "C" void kernel_launch(void* const* d_in, const int* in_sizes, int n_in,
                              void* d_out, int out_size, void* d_ws, size_t ws_size,
                              hipStream_t stream) {
  const float* JtJ    = (const float*)d_in[0];
  const float* Jt     = (const float*)d_in[1];
  const float* wts    = (const float*)d_in[2];
  const float* Rres   = (const float*)d_in[3];
  const float* poseR  = (const float*)d_in[4];
  const float* poset  = (const float*)d_in[5];
  // d_in[6..10]: invD0, invD1, x0, x1, K -> unused by reference output
  const float* consts = (const float*)d_in[11];
  const float* c1w    = (const float*)d_in[12];
  const float* c1b    = (const float*)d_in[13];
  const float* c2w    = (const float*)d_in[14];
  const float* c2b    = (const float*)d_in[15];
  const float* f1w    = (const float*)d_in[16];
  const float* f1b    = (const float*)d_in[17];
  const float* f2w    = (const float*)d_in[18];
  const float* f2b    = (const float*)d_in[19];
  float* out = (float*)d_out;

  // workspace layout (floats), total ~20.3 MB
  float* ws      = (float*)d_ws;
  float* p1      = ws;                             // 1,228,800
  float* a1      = p1 + (size_t)BATCH*P1H*P1W;     // 2,457,600
  float* hbuf    = a1 + (size_t)BATCH*C1*A1H*A1W;  // 1,228,800
  float* partial = hbuf + (size_t)BATCH*KFC;       // 122,880
  float* Hfc     = partial + (size_t)KSPLIT*64*64; // 4,096
  float* lamb    = Hfc + 64*64;                    // 384
  float* jtrp    = lamb + BATCH*6;                 // 24,576

  // DampingNet pipeline
  {
    int n = BATCH*P1H*P1W;
    k_pool0<<<(n+255)/256, 256, 0, stream>>>(Rres, p1);
  }
  {
    int n = BATCH*C1*A1H*A1W;
    k_conv1pool<<<(n+255)/256, 256, 0, stream>>>(p1, c1w, c1b, a1);
  }
  {
    int n = BATCH*C2*A2H*A2W;
    k_conv2pool<<<(n+255)/256, 256, 0, stream>>>(a1, c2w, c2b, hbuf);
  }
  k_fc1_wmma<<<KSPLIT, 512, 0, stream>>>(hbuf, f1w, partial);
  k_fc1_reduce<<<16, 256, 0, stream>>>(partial, f1b, Hfc);
  k_damp<<<1, 64, 0, stream>>>(Hfc, f2w, f2b, consts, lamb);

  // JtR (bandwidth-dominant term), B128 loads, deterministic two-phase reduction
  k_jtr_partial<<<dim3(BATCH, NCHUNK), 256, 0, stream>>>(
      (const float4*)Jt, (const float4*)wts, (const float4*)Rres, jtrp);

  // 6x6 solve + Rodrigues + pose composition
  k_solve<<<1, 64, 0, stream>>>(JtJ, jtrp, lamb, poseR, poset, out);
}